// GCN_57166014710279
// MI455X (gfx1250) — compile-verified
//
#include <hip/hip_runtime.h>
#include <hip/hip_bf16.h>

// GCN layer for MI455X (gfx1250, wave32).
//   agg = scatter_add(h[src], dst);  deg = bincount(src)
//   out = relu((agg @ W) * rsqrt(deg)[:,None] + bias)
//
// d_ws layout: [ agg : n_nodes*128 f32 ][ deg : n_nodes u32 ]  (~20.6 MB)

#define D_FEAT 128
#define SA_PITCH (D_FEAT + 4)   // LDS A-strip pitch: 132 floats -> conflict-free TR reads

typedef __attribute__((ext_vector_type(2))) float v2f;
typedef __attribute__((ext_vector_type(8))) float v8f;

// ---------------------------------------------------------------------------
// Kernel 1: zero the workspace (accumulator + degree counts).
// ---------------------------------------------------------------------------
__global__ __launch_bounds__(256) void gcn_zero_ws(float4* __restrict__ agg4,
                                                   unsigned* __restrict__ deg,
                                                   int n_agg4, int n_nodes) {
  const int i = blockIdx.x * blockDim.x + threadIdx.x;
  if (i < n_agg4) agg4[i] = make_float4(0.f, 0.f, 0.f, 0.f);
  if (i < n_nodes) deg[i] = 0u;
}

// ---------------------------------------------------------------------------
// Kernel 2: edge scatter. One wave32 per edge: 32 lanes x float4 = 128 floats.
// h row comes in as global_load_b128; accumulation is hardware
// global_atomic_add_f32 (relaxed / agent scope, no return) into L2-resident agg.
// ---------------------------------------------------------------------------
__global__ __launch_bounds__(256) void gcn_scatter(const float* __restrict__ h,
                                                   const long long* __restrict__ src,
                                                   const long long* __restrict__ dst,
                                                   float* __restrict__ agg,
                                                   unsigned* __restrict__ deg,
                                                   int n_edges) {
  const int e = blockIdx.x * 8 + (threadIdx.x >> 5);   // 8 waves per block
  if (e >= n_edges) return;
  const int lane = threadIdx.x & 31;

  const long long s = src[e];   // wave-uniform
  const long long d = dst[e];   // wave-uniform

  const float4 v = *(const float4*)(h + (size_t)s * D_FEAT + (size_t)lane * 4);
  float* arow = agg + (size_t)d * D_FEAT + (size_t)lane * 4;

  __hip_atomic_fetch_add(&arow[0], v.x, __ATOMIC_RELAXED, __HIP_MEMORY_SCOPE_AGENT);
  __hip_atomic_fetch_add(&arow[1], v.y, __ATOMIC_RELAXED, __HIP_MEMORY_SCOPE_AGENT);
  __hip_atomic_fetch_add(&arow[2], v.z, __ATOMIC_RELAXED, __HIP_MEMORY_SCOPE_AGENT);
  __hip_atomic_fetch_add(&arow[3], v.w, __ATOMIC_RELAXED, __HIP_MEMORY_SCOPE_AGENT);

  if (lane == 0) {
    __hip_atomic_fetch_add(&deg[s], 1u, __ATOMIC_RELAXED, __HIP_MEMORY_SCOPE_AGENT);
  }
}

// ---------------------------------------------------------------------------
// Kernel 3: dense transform + fused epilogue via V_WMMA_F32_16X16X4_F32.
// Block = 256 threads = 8 waves; block owns a 16-row strip, each wave one
// 16x16 output tile (16 cols). W (64 KB) + A strip (padded) staged in LDS.
// ---------------------------------------------------------------------------
__global__ __launch_bounds__(256) void gcn_gemm_epilogue(
    const float* __restrict__ agg, const unsigned* __restrict__ deg,
    const float* __restrict__ W, const float* __restrict__ bias,
    float* __restrict__ out, int n_nodes) {
  __shared__ float sW[D_FEAT * D_FEAT];     // 64 KB, row-major [k][n]
  __shared__ float sA[16 * SA_PITCH];       // 16 rows x 132, padded

  const int t = threadIdx.x;
  const int rowBase = blockIdx.x * 16;

  // Cooperative stage of W: 16384 floats / 256 threads = 16 float4 each.
  for (int i = t * 4; i < D_FEAT * D_FEAT; i += 256 * 4) {
    *(float4*)&sW[i] = *(const float4*)&W[i];
  }
  // Stage 16x128 A-strip (2048 floats): 2 float4 per thread, padded pitch.
  for (int i = t * 4; i < 16 * D_FEAT; i += 256 * 4) {
    const int r = i >> 7;            // /128
    const int c = i & (D_FEAT - 1);  // %128
    const int row = rowBase + r;
    float4 v = make_float4(0.f, 0.f, 0.f, 0.f);
    if (row < n_nodes) v = *(const float4*)&agg[(size_t)row * D_FEAT + c];
    *(float4*)&sA[r * SA_PITCH + c] = v;
  }
  __syncthreads();

  const int lane = t & 31;
  const int colBase = (t >> 5) * 16;   // wave id -> 16-col tile
  const int half = lane >> 4;          // 0: K+{0,1}, 1: K+{2,3}
  const int l = lane & 15;             // A: row M=l ; B: col N=l

  v8f acc = {};   // 16x16 f32 C/D fragment (8 VGPRs)

  // K = 128 as 32 steps of V_WMMA_F32_16X16X4_F32.
  // A fragment (16x4): lanes 0-15 hold K=k,k+1 of row l; lanes 16-31 K=k+2,k+3.
  // B fragment (4x16): lanes 0-15 hold K=k,k+1 of col l; lanes 16-31 K=k+2,k+3.
#pragma unroll
  for (int k = 0; k < D_FEAT; k += 4) {
    const int ka = k + half * 2;
    v2f a, b;
    a[0] = sA[l * SA_PITCH + ka];
    a[1] = sA[l * SA_PITCH + ka + 1];
    b[0] = sW[ka * D_FEAT + colBase + l];
    b[1] = sW[(ka + 1) * D_FEAT + colBase + l];
    acc = __builtin_amdgcn_wmma_f32_16x16x4_f32(
        /*neg_a=*/false, a, /*neg_b=*/false, b,
        /*c_mod=*/(short)0, acc, /*reuse_a=*/false, /*reuse_b=*/false);
  }

  // Fused epilogue: out[row, col] = relu(acc * rsqrt(deg[row]) + bias[col]).
  // C/D layout: VGPR r, lanes 0-15 -> M=r, lanes 16-31 -> M=r+8; N=l.
  const float bcol = bias[colBase + l];
#pragma unroll
  for (int r = 0; r < 8; ++r) {
    const int row = rowBase + r + half * 8;
    if (row < n_nodes) {
      const float nrm = rsqrtf((float)deg[row]);   // deg==0 -> inf, matches ref
      const float v = fmaf(acc[r], nrm, bcol);
      out[(size_t)row * D_FEAT + colBase + l] = fmaxf(v, 0.f);
    }
  }
}

// ---------------------------------------------------------------------------
extern "C" void kernel_launch(void* const* d_in, const int* in_sizes, int n_in,
                              void* d_out, int out_size, void* d_ws, size_t ws_size,
                              hipStream_t stream) {
  const float* h        = (const float*)d_in[0];
  const long long* src  = (const long long*)d_in[1];   // jnp.int64
  const long long* dst  = (const long long*)d_in[2];   // jnp.int64
  const float* W        = (const float*)d_in[3];
  const float* bias     = (const float*)d_in[4];
  float* out            = (float*)d_out;

  const int n_nodes = in_sizes[0] / D_FEAT;
  const int n_edges = in_sizes[1];

  float* agg    = (float*)d_ws;
  unsigned* deg = (unsigned*)((char*)d_ws + (size_t)n_nodes * D_FEAT * sizeof(float));

  // 1) zero accumulator + degree counts
  const int n_agg4 = n_nodes * D_FEAT / 4;
  {
    const int threads = 256;
    const int blocks = (n_agg4 + threads - 1) / threads;   // covers n_nodes too
    gcn_zero_ws<<<blocks, threads, 0, stream>>>((float4*)agg, deg, n_agg4, n_nodes);
  }

  // 2) edge scatter: one wave32 per edge, 8 edges per 256-thread block
  {
    const int blocks = (n_edges + 7) / 8;
    gcn_scatter<<<blocks, 256, 0, stream>>>(h, src, dst, agg, deg, n_edges);
  }

  // 3) WMMA GEMM + fused norm/bias/relu: one block per 16-row strip
  {
    const int blocks = (n_nodes + 15) / 16;
    gcn_gemm_epilogue<<<blocks, 256, 0, stream>>>(agg, deg, W, bias, out, n_nodes);
  }
}